// mmd_loss_12455405159029
// MI455X (gfx1250) — compile-verified
//
#include <hip/hip_runtime.h>
#include <math.h>

typedef __attribute__((ext_vector_type(2))) float v2f;
typedef __attribute__((ext_vector_type(8))) float v8f;

// ---------------------------------------------------------------------------
// ws layout: [0,24)   : 3 doubles  -> sum(l2) for pairs {ss, tt, st}
//            [32, ...): 2N floats  -> row squared norms (S rows, then T rows)
// ---------------------------------------------------------------------------

__global__ void mmd_zero_kernel(double* sums, float* out) {
    if (threadIdx.x < 3) sums[threadIdx.x] = 0.0;
    if (threadIdx.x == 3) out[0] = 0.0f;
}

__global__ void mmd_sqnorm_kernel(const float* __restrict__ S,
                                  const float* __restrict__ T,
                                  float* __restrict__ sq, int N, int D) {
    int i = blockIdx.x * blockDim.x + threadIdx.x;
    if (i >= 2 * N) return;
    const float* row = (i < N) ? (S + (size_t)i * D) : (T + (size_t)(i - N) * D);
    const float4* r4 = (const float4*)row;
    float s = 0.f;
    for (int k = 0; k < D / 4; ++k) {
        float4 v = r4[k];
        s += v.x * v.x + v.y * v.y + v.z * v.z + v.w * v.w;
    }
    sq[i] = s;
}

// One wave computes a 32x32 tile of G = A * B^T as 2x2 WMMA accumulators.
// fp32 WMMA 16x16x4: A fragment = float2 (K pair); B^T fragment uses the same
// contiguous-float2 addressing pattern on the row-major operand.
__device__ __forceinline__ void gemm_tile32(const float* __restrict__ A,
                                            const float* __restrict__ B,
                                            int D, int row0, int col0,
                                            v8f acc[4]) {
    const int lane = threadIdx.x & 31;
    const int m  = lane & 15;
    const int kq = (lane >> 4) << 1;        // 0 for lanes 0-15, 2 for 16-31
    const float* a0p = A + (size_t)(row0 + m) * D + kq;
    const float* a1p = a0p + (size_t)16 * D;
    const float* b0p = B + (size_t)(col0 + m) * D + kq;
    const float* b1p = b0p + (size_t)16 * D;

    const v8f z = {0.f, 0.f, 0.f, 0.f, 0.f, 0.f, 0.f, 0.f};
    acc[0] = z; acc[1] = z; acc[2] = z; acc[3] = z;

    for (int k = 0; k < D; k += 4) {
        v2f a0 = *(const v2f*)(a0p + k);
        v2f a1 = *(const v2f*)(a1p + k);
        v2f b0 = *(const v2f*)(b0p + k);
        v2f b1 = *(const v2f*)(b1p + k);
        acc[0] = __builtin_amdgcn_wmma_f32_16x16x4_f32(false, a0, false, b0,
                                                       (short)0, acc[0], false, false);
        acc[1] = __builtin_amdgcn_wmma_f32_16x16x4_f32(false, a0, false, b1,
                                                       (short)0, acc[1], false, false);
        acc[2] = __builtin_amdgcn_wmma_f32_16x16x4_f32(false, a1, false, b0,
                                                       (short)0, acc[2], false, false);
        acc[3] = __builtin_amdgcn_wmma_f32_16x16x4_f32(false, a1, false, b1,
                                                       (short)0, acc[3], false, false);
    }
}

__device__ __forceinline__ void pair_setup(int p, const float* S, const float* T,
                                           const float* sq, int N,
                                           const float*& A, const float*& B,
                                           const float*& sqa, const float*& sqb) {
    A   = (p == 1) ? T : S;
    B   = (p == 0) ? S : T;
    sqa = (p == 1) ? (sq + N) : sq;
    sqb = (p == 0) ? sq : (sq + N);
}

// 5-term Gaussian ladder: bandwidths bw*2^i, i=0..4.  With
// e = exp(-l2/(16*bw)):  sum_i exp(-l2/(bw*2^i)) = e + e^2 + e^4 + e^8 + e^16.
// 1 transcendental + 4 squarings instead of 5 transcendentals.
__device__ __forceinline__ float kernel5(float l2, float neg_inv_bw16) {
    float e  = __expf(l2 * neg_inv_bw16);
    float e2 = e * e;
    float e4 = e2 * e2;
    float e8 = e4 * e4;
    float e16 = e8 * e8;
    return e + e2 + e4 + e8 + e16;
}

// Pass 1: accumulate sum(l2) per pair (needed globally for the bandwidths).
// ss/tt are symmetric: only col_tile >= row_tile computed, strict-upper x2.
__global__ void mmd_l2sum_kernel(const float* __restrict__ S,
                                 const float* __restrict__ T,
                                 const float* __restrict__ sq,
                                 double* __restrict__ sums, int N, int D) {
    const int p = blockIdx.z;
    const int row0 = blockIdx.y * 32;
    const int col0 = blockIdx.x * 32;
    if (p != 2 && col0 < row0) return;          // symmetric pair: upper only
    const float wt = (p != 2 && col0 > row0) ? 2.0f : 1.0f;

    const float *A, *B, *sqa, *sqb;
    pair_setup(p, S, T, sq, N, A, B, sqa, sqb);

    v8f acc[4];
    gemm_tile32(A, B, D, row0, col0, acc);

    const int lane = threadIdx.x & 31;
    const int mh = (lane >> 4) * 8;   // row half offset
    const int nn = lane & 15;         // column within 16

    float local = 0.f;
#pragma unroll
    for (int t = 0; t < 4; ++t) {
        const int rb = row0 + ((t >> 1) * 16) + mh;
        const int cb = col0 + ((t & 1) * 16) + nn;
        const float sb = sqb[cb];
#pragma unroll
        for (int v = 0; v < 8; ++v)
            local += sqa[rb + v] + sb - 2.0f * acc[t][v];
    }
    double dl = (double)(local * wt);
#pragma unroll
    for (int off = 16; off > 0; off >>= 1)
        dl += __shfl_xor(dl, off, 32);
    if (lane == 0) atomicAdd(&sums[p], dl);
}

// Pass 2: recompute Gram tiles, evaluate the Gaussian kernel sums with the
// proper bandwidth(s) per pair, reduce to the scalar MMD.
__global__ void mmd_final_kernel(const float* __restrict__ S,
                                 const float* __restrict__ T,
                                 const float* __restrict__ sq,
                                 const double* __restrict__ sums,
                                 float* __restrict__ out, int N, int D) {
    const int p = blockIdx.z;
    const int row0 = blockIdx.y * 32;
    const int col0 = blockIdx.x * 32;
    if (p != 2 && col0 < row0) return;          // symmetric pair: upper only
    const float wt = (p != 2 && col0 > row0) ? 2.0f : 1.0f;

    const float *A, *B, *sqa, *sqb;
    pair_setup(p, S, T, sq, N, A, B, sqa, sqb);

    const double m  = 2.0 * (double)N;             // 4096
    const double mm = m * m - m;                   // M^2 - M
    const double Sss = sums[0], Stt = sums[1], Sst = sums[2];
    // bw_s = (4*Sss/mm)/4 = Sss/mm  (block replication cancels /KERNEL_MUL^2)
    const float bw_own = (p == 0) ? (float)(Sss / mm) : (float)(Stt / mm);
    const float bw_c   = (float)((Sss + Stt + 2.0 * Sst) / mm * 0.25);
    const float w_own  = (float)(1.0 / ((double)N * (double)N));
    const float w_c    = (p == 2) ? (float)(-4.0 / (m * m))
                                  : (float)(-2.0 / (m * m));
    const float ci_own = -1.0f / (16.0f * bw_own);
    const float ci_c   = -1.0f / (16.0f * bw_c);

    v8f acc[4];
    gemm_tile32(A, B, D, row0, col0, acc);

    const int lane = threadIdx.x & 31;
    const int mh = (lane >> 4) * 8;
    const int nn = lane & 15;

    float local = 0.f;
#pragma unroll
    for (int t = 0; t < 4; ++t) {
        const int rb = row0 + ((t >> 1) * 16) + mh;
        const int cb = col0 + ((t & 1) * 16) + nn;
        const float sb = sqb[cb];
#pragma unroll
        for (int v = 0; v < 8; ++v) {
            const float l2 = sqa[rb + v] + sb - 2.0f * acc[t][v];
            local += w_c * kernel5(l2, ci_c);
            if (p != 2) local += w_own * kernel5(l2, ci_own);
        }
    }
    local *= wt;
#pragma unroll
    for (int off = 16; off > 0; off >>= 1)
        local += __shfl_xor(local, off, 32);
    if (lane == 0) atomicAdd(out, local);
}

extern "C" void kernel_launch(void* const* d_in, const int* in_sizes, int n_in,
                              void* d_out, int out_size, void* d_ws, size_t ws_size,
                              hipStream_t stream) {
    const float* S = (const float*)d_in[0];
    const float* T = (const float*)d_in[1];
    float* out = (float*)d_out;
    const int D = 256;                 // feature dim from reference
    const int N = in_sizes[0] / D;     // 2048 rows per input

    double* sums = (double*)d_ws;                       // 3 doubles
    float*  sq   = (float*)((char*)d_ws + 32);          // 2N floats

    mmd_zero_kernel<<<1, 32, 0, stream>>>(sums, out);
    mmd_sqnorm_kernel<<<(2 * N + 255) / 256, 256, 0, stream>>>(S, T, sq, N, D);

    dim3 grid(N / 32, N / 32, 3);
    mmd_l2sum_kernel<<<grid, 32, 0, stream>>>(S, T, sq, sums, N, D);
    mmd_final_kernel<<<grid, 32, 0, stream>>>(S, T, sq, sums, out, N, D);
}